// MHLP_layer_20813411516919
// MI455X (gfx1250) — compile-verified
//
#include <hip/hip_runtime.h>
#include <climits>

// Problem constants (match reference)
#define NN   100000
#define EE   100000
#define LL   4
#define DD   64
#define HH   4
#define MM   4
#define TT   8192
#define AA   128
#define OUTD 64
#define HDD  256          // H*D
static constexpr float ALPHA = 0.01f;

typedef __attribute__((ext_vector_type(2))) float v2f;
typedef __attribute__((ext_vector_type(8))) float v8f;

// ---- ordered-int encoding for float atomicMax ----
__device__ __forceinline__ int f2ord(float f) {
  int i = __float_as_int(f);
  return i >= 0 ? i : (i ^ 0x7fffffff);
}
__device__ __forceinline__ float ord2f(int i) {
  return __int_as_float(i >= 0 ? i : (i ^ 0x7fffffff));
}

// =====================================================================
// f32 WMMA GEMM with LDS-staged A tile and software-pipelined operand
// fetch:  C[Mrows x N] = A[Mrows x K] * W[N x K]^T + bias
// Block = 128 threads (4 waves).  Each block owns a 16-row M-tile; the
// 16 x K A-tile is cooperatively staged into LDS (row stride K+4 floats
// -> float4-aligned stores, 4*row bank spread over 64 banks).
// Each wave owns NTPW consecutive 16-col n-tiles.  Per K-step: one LDS
// A-fragment (ds_load_[2addr_]b64) + NTPW global b64 B-fragments feed
// NTPW back-to-back v_wmma_f32_16x16x4_f32; the NEXT step's fragments
// are preloaded before the WMMA chain so loads overlap matrix issue.
// Requires: N == 4*NTPW*16; K % 4 == 0; Mrows % 16 == 0 (E, T qualify).
// Fragment packing per ISA 7.12.2 (32-bit 16x4): lanes 0-15 hold K=0,1
// in v[0],v[1]; lanes 16-31 hold K=2,3.  B mirrored (col per lane).
// =====================================================================
template <int NTPW>
__global__ void wmma_gemm_f32_lds(const float* __restrict__ A,
                                  const float* __restrict__ W,
                                  const float* __restrict__ bias,
                                  float* __restrict__ C,
                                  int Mrows, int N, int K) {
  extern __shared__ float lds[];            // 16 * (K+4) floats
  const int lane = threadIdx.x & 31;
  const int wave = threadIdx.x >> 5;
  const int row0 = blockIdx.x * 16;
  const int kstride = K + 4;                // LDS row stride (banking + alignment)

  // ---- cooperative stage of A tile (16 x K) into LDS ----
  {
    const float4* As = (const float4*)(A + (size_t)row0 * K);
    const int nvec = 16 * K / 4;
    const int kv = K / 4;
    for (int v = threadIdx.x; v < nvec; v += blockDim.x) {
      int r = v / kv, c = v - r * kv;
      *(float4*)&lds[r * kstride + 4 * c] = As[v];
    }
  }
  __syncthreads();

  const int kb = (lane >> 4) << 1;          // 0 or 2
  int col[NTPW];
  const float* wp[NTPW];
  v8f acc[NTPW];
#pragma unroll
  for (int i = 0; i < NTPW; ++i) {
    col[i] = (wave * NTPW + i) * 16 + (lane & 15);
    wp[i] = W + (size_t)col[i] * K + kb;
    __builtin_prefetch(wp[i], 0, 1);        // global_prefetch_b8
    const float b = bias ? bias[col[i]] : 0.0f;
#pragma unroll
    for (int j = 0; j < 8; ++j) acc[i][j] = b;
  }

  const float* arow = lds + (lane & 15) * kstride + kb;

  // prologue: fetch k=0 fragments
  v2f af = *(const v2f*)(arow);
  v2f bf[NTPW];
#pragma unroll
  for (int i = 0; i < NTPW; ++i) bf[i] = *(const v2f*)(wp[i]);

  for (int k = 0; k < K; k += 4) {
    // preload next step's fragments (loads issue before the WMMA chain)
    v2f af_n = af;
    v2f bf_n[NTPW];
#pragma unroll
    for (int i = 0; i < NTPW; ++i) bf_n[i] = bf[i];
    const int kn = k + 4;
    if (kn < K) {
      af_n = *(const v2f*)(arow + kn);
#pragma unroll
      for (int i = 0; i < NTPW; ++i) bf_n[i] = *(const v2f*)(wp[i] + kn);
    }
    // matrix issue for current step
#pragma unroll
    for (int i = 0; i < NTPW; ++i) {
      acc[i] = __builtin_amdgcn_wmma_f32_16x16x4_f32(
          /*neg_a=*/false, af, /*neg_b=*/false, bf[i],
          /*c_mod=*/(short)0, acc[i], /*reuse_a=*/false, /*reuse_b=*/false);
    }
    af = af_n;
#pragma unroll
    for (int i = 0; i < NTPW; ++i) bf[i] = bf_n[i];
  }

  // C/D layout: VGPR j -> row = row0 + j + 8*(lane>>4)
  const int rb = row0 + ((lane >> 4) << 3);
#pragma unroll
  for (int i = 0; i < NTPW; ++i) {
#pragma unroll
    for (int j = 0; j < 8; ++j) {
      int r = rb + j;
      C[(size_t)r * N + col[i]] = acc[i][j];
    }
  }
}

// ---------------- per-metapath helper kernels ----------------

// reset map / mx / ssum / ftslot for one metapath
__global__ void k_init_mp(int* __restrict__ map, int* __restrict__ mxb,
                          float* __restrict__ ssum, float* __restrict__ ftslot) {
  int i = blockIdx.x * blockDim.x + threadIdx.x;
  if (i < NN)        map[i] = INT_MAX;
  if (i < TT * HH) { mxb[i] = INT_MIN; ssum[i] = 0.f; }
  if (i < TT * HDD)  ftslot[i] = 0.f;
}

// node -> output slot (first occurrence in tgt)
__global__ void k_map(const int* __restrict__ tgt_m, int* __restrict__ map) {
  int t = blockIdx.x * blockDim.x + threadIdx.x;
  if (t < TT) atomicMin(&map[tgt_m[t]], t);
}

// agg[e][d] = mean_l features[emi[e][l]][d]
__global__ void k_agg(const float* __restrict__ features,
                      const int* __restrict__ emi_m,
                      float* __restrict__ agg) {
  int i = blockIdx.x * blockDim.x + threadIdx.x;
  if (i >= EE * DD) return;
  int e = i >> 6, d = i & 63;
  const int* ep = emi_m + e * LL;
  float s = 0.f;
#pragma unroll
  for (int l = 0; l < LL; ++l) s += features[(size_t)ep[l] * DD + d];
  agg[i] = s * 0.25f;
}

// att[e][h] = leaky( center . Wattn1[h] + eft[e,h,:] . attn2[h] )
__global__ void k_att(const float* __restrict__ features,
                      const int* __restrict__ emi_m,
                      const float* __restrict__ eft,
                      const float* __restrict__ Wattn1_m,
                      const float* __restrict__ attn2_m,
                      float* __restrict__ attb) {
  int i = blockIdx.x * blockDim.x + threadIdx.x;
  if (i >= EE * HH) return;
  int e = i >> 2, h = i & 3;
  int ci = emi_m[e * LL + (LL - 1)];
  const float* c  = features + (size_t)ci * DD;
  const float* w1 = Wattn1_m + h * DD;
  const float* a2 = attn2_m + h * DD;
  const float* ef = eft + (size_t)e * HDD + h * DD;
  float s = 0.f;
#pragma unroll 8
  for (int d = 0; d < DD; ++d) s += c[d] * w1[d] + ef[d] * a2[d];
  attb[i] = s > 0.f ? s : ALPHA * s;
}

// segment max (only for targeted dst nodes)
__global__ void k_max(const int* __restrict__ dst_m, const int* __restrict__ map,
                      const float* __restrict__ attb, int* __restrict__ mxb) {
  int i = blockIdx.x * blockDim.x + threadIdx.x;
  if (i >= EE * HH) return;
  int e = i >> 2, h = i & 3;
  int slot = map[dst_m[e]];
  if (slot == INT_MAX) return;
  atomicMax(&mxb[slot * HH + h], f2ord(attb[i]));
}

// ex = exp(att - mx[dst]); segment sum of ex
__global__ void k_exp(const int* __restrict__ dst_m, const int* __restrict__ map,
                      const float* __restrict__ attb, const int* __restrict__ mxb,
                      float* __restrict__ exb, float* __restrict__ ssum) {
  int i = blockIdx.x * blockDim.x + threadIdx.x;
  if (i >= EE * HH) return;
  int e = i >> 2, h = i & 3;
  int slot = map[dst_m[e]];
  float v = 0.f;
  if (slot != INT_MAX) {
    v = __expf(attb[i] - ord2f(mxb[slot * HH + h]));
    atomicAdd(&ssum[slot * HH + h], v);
  }
  exb[i] = v;
}

// ftslot[slot][d] += eft[e][d] * a[e][h]
__global__ void k_scatter(const int* __restrict__ dst_m, const int* __restrict__ map,
                          const float* __restrict__ eft, const float* __restrict__ exb,
                          const float* __restrict__ ssum, float* __restrict__ ftslot) {
  int i = blockIdx.x * blockDim.x + threadIdx.x;
  if (i >= EE * HDD) return;
  int e = i >> 8, d = i & 255;
  int slot = map[dst_m[e]];
  if (slot == INT_MAX) return;
  int h = d >> 6;
  float a = exb[e * HH + h] / ssum[slot * HH + h];
  atomicAdd(&ftslot[(size_t)slot * HDD + d], eft[(size_t)e * HDD + d] * a);
}

// mo[m][t][d] = elu(ftslot[map[tgt[t]]][d])
__global__ void k_moout(const int* __restrict__ tgt_m, const int* __restrict__ map,
                        const float* __restrict__ ftslot, float* __restrict__ mo_m) {
  int i = blockIdx.x * blockDim.x + threadIdx.x;
  if (i >= TT * HDD) return;
  int t = i >> 8, d = i & 255;
  int slot = map[tgt_m[t]];
  float v = (slot != INT_MAX) ? ftslot[(size_t)slot * HDD + d] : 0.f;
  mo_m[i] = v > 0.f ? v : __expf(v) - 1.f;
}

// ---------------- inter-path kernels ----------------

// colsum[a] = sum_t tanh(tmp[t][a])   (block per column, LDS tree reduce)
__global__ void k_redtanh(const float* __restrict__ tmp, float* __restrict__ colsum) {
  __shared__ float sh[256];
  int col = blockIdx.x;
  float s = 0.f;
  for (int t = threadIdx.x; t < TT; t += blockDim.x)
    s += tanhf(tmp[(size_t)t * AA + col]);
  sh[threadIdx.x] = s;
  __syncthreads();
  for (int o = 128; o > 0; o >>= 1) {
    if ((int)threadIdx.x < o) sh[threadIdx.x] += sh[threadIdx.x + o];
    __syncthreads();
  }
  if (threadIdx.x == 0) colsum[col] = sh[0];
}

// sval = dot(colsum, w2) / T
__global__ void k_sdot(const float* __restrict__ colsum, const float* __restrict__ w2,
                       float* __restrict__ sval) {
  __shared__ float sh[AA];
  int i = threadIdx.x;
  sh[i] = colsum[i] * w2[i];
  __syncthreads();
  for (int o = AA / 2; o > 0; o >>= 1) {
    if (i < o) sh[i] += sh[i + o];
    __syncthreads();
  }
  if (i == 0) *sval = sh[0] / (float)TT;
}

// h = softmax(s)[0]*mo0 + softmax(s)[1]*mo1
__global__ void k_combine(const float* __restrict__ svals, const float* __restrict__ mo0,
                          const float* __restrict__ mo1, float* __restrict__ hbuf) {
  int i = blockIdx.x * blockDim.x + threadIdx.x;
  if (i >= TT * HDD) return;
  float s0 = svals[0], s1 = svals[1];
  float mx = fmaxf(s0, s1);
  float e0 = __expf(s0 - mx), e1 = __expf(s1 - mx);
  float w0 = e0 / (e0 + e1);
  hbuf[i] = w0 * mo0[i] + (1.f - w0) * mo1[i];
}

// =====================================================================
extern "C" void kernel_launch(void* const* d_in, const int* in_sizes, int n_in,
                              void* d_out, int out_size, void* d_ws, size_t ws_size,
                              hipStream_t stream) {
  (void)in_sizes; (void)n_in; (void)out_size; (void)ws_size;

  const float* features = (const float*)d_in[0];   // N x D
  const float* Wexp     = (const float*)d_in[1];   // M x HD x D
  const float* bexp     = (const float*)d_in[2];   // M x HD
  const float* Wattn1   = (const float*)d_in[3];   // M x H x D
  const float* attn2    = (const float*)d_in[4];   // M x H x D
  const float* Wfc1     = (const float*)d_in[5];   // 2 x A x HD
  const float* bfc1     = (const float*)d_in[6];   // 2 x A
  const float* wfc2     = (const float*)d_in[7];   // 2 x A
  const float* Wout     = (const float*)d_in[8];   // 2 x OUT x HD
  const float* bout     = (const float*)d_in[9];   // 2 x OUT
  const int*   emi      = (const int*)d_in[10];    // M x E x L
  const int*   dst      = (const int*)d_in[11];    // M x E
  const int*   tgt      = (const int*)d_in[12];    // M x T
  float* out = (float*)d_out;                      // 2 x T x OUT

  // ---- workspace carve-up ----
  float* ws = (float*)d_ws;
  size_t off = 0;
  float* agg    = ws + off; off += (size_t)EE * DD;        // 6.4M
  float* eft    = ws + off; off += (size_t)EE * HDD;       // 25.6M
  float* attb   = ws + off; off += (size_t)EE * HH;
  float* exb    = ws + off; off += (size_t)EE * HH;
  float* ftslot = ws + off; off += (size_t)TT * HDD;
  float* mo     = ws + off; off += (size_t)MM * TT * HDD;
  float* tmp    = ws + off; off += (size_t)TT * AA;
  float* colsum = ws + off; off += AA;
  float* svals  = ws + off; off += 2;
  float* ssum   = ws + off; off += (size_t)TT * HH;
  float* hbuf   = ws + off; off += (size_t)TT * HDD;
  int* map = (int*)(ws + off);
  int* mxb = map + NN;

  const int B = 256;
  auto cdiv = [](long long a, long long b) { return (int)((a + b - 1) / b); };
  auto shb  = [](int K) { return (size_t)(16 * (K + 4)) * sizeof(float); };

  // ---- per-metapath intra aggregation ----
  for (int m = 0; m < MM; ++m) {
    const int* emi_m = emi + (size_t)m * EE * LL;
    const int* dst_m = dst + (size_t)m * EE;
    const int* tgt_m = tgt + (size_t)m * TT;
    float* mo_m = mo + (size_t)m * TT * HDD;

    k_init_mp<<<cdiv((long long)TT * HDD, B), B, 0, stream>>>(map, mxb, ssum, ftslot);
    k_map<<<cdiv(TT, B), B, 0, stream>>>(tgt_m, map);
    k_agg<<<cdiv((long long)EE * DD, B), B, 0, stream>>>(features, emi_m, agg);
    // eft = agg @ Wexp^T + bexp   (E x 64) x (64 x 256): 4 waves x 4 n-tiles
    wmma_gemm_f32_lds<4><<<EE / 16, 128, shb(DD), stream>>>(
        agg, Wexp + (size_t)m * HDD * DD, bexp + (size_t)m * HDD, eft, EE, HDD, DD);
    k_att<<<cdiv((long long)EE * HH, B), B, 0, stream>>>(
        features, emi_m, eft, Wattn1 + (size_t)m * HH * DD,
        attn2 + (size_t)m * HH * DD, attb);
    k_max<<<cdiv((long long)EE * HH, B), B, 0, stream>>>(dst_m, map, attb, mxb);
    k_exp<<<cdiv((long long)EE * HH, B), B, 0, stream>>>(dst_m, map, attb, mxb, exb, ssum);
    k_scatter<<<cdiv((long long)EE * HDD, B), B, 0, stream>>>(dst_m, map, eft, exb, ssum, ftslot);
    k_moout<<<cdiv((long long)TT * HDD, B), B, 0, stream>>>(tgt_m, map, ftslot, mo_m);
  }

  // ---- inter-path attention + output heads ----
  for (int g = 0; g < 2; ++g) {
    for (int j = 0; j < 2; ++j) {
      const float* mo_m = mo + (size_t)(2 * g + j) * TT * HDD;
      // tmp = mo_m @ Wfc1[g]^T + bfc1[g]   (T x 256) x (256 x 128): 4 waves x 2
      wmma_gemm_f32_lds<2><<<TT / 16, 128, shb(HDD), stream>>>(
          mo_m, Wfc1 + (size_t)g * AA * HDD, bfc1 + (size_t)g * AA, tmp, TT, AA, HDD);
      k_redtanh<<<AA, 256, 0, stream>>>(tmp, colsum);
      k_sdot<<<1, AA, 0, stream>>>(colsum, wfc2 + (size_t)g * AA, svals + j);
    }
    k_combine<<<cdiv((long long)TT * HDD, B), B, 0, stream>>>(
        svals, mo + (size_t)(2 * g) * TT * HDD, mo + (size_t)(2 * g + 1) * TT * HDD, hbuf);
    // logits = hbuf @ Wout[g]^T + bout[g]   (T x 256) x (256 x 64): 4 waves x 1
    wmma_gemm_f32_lds<1><<<TT / 16, 128, shb(HDD), stream>>>(
        hbuf, Wout + (size_t)g * OUTD * HDD, bout + (size_t)g * OUTD,
        out + (size_t)g * TT * OUTD, TT, OUTD, HDD);
  }
}